// YOLOLoss_72155450573407
// MI455X (gfx1250) — compile-verified
//
#include <hip/hip_runtime.h>

// ---------------------------------------------------------------------------
// YOLO loss, MI455X (gfx1250, wave32).
// total_obj/cls = sum over scales of mean(softplus(x) - x*z)
//   = [sum softplus(x) - sum_{z=1} x] / count     (z sparse: <=200 cells)
// ---------------------------------------------------------------------------

typedef float v2f __attribute__((ext_vector_type(2)));
typedef float v4f __attribute__((ext_vector_type(4)));
typedef float v8f __attribute__((ext_vector_type(8)));

#define YB 16
#define YA 3
#define YC 80
#define YCH 85   // 5 + C

__device__ __forceinline__ float softplus_f(float x) {
  // log(1+exp(x)) = max(x,0) + log1p(exp(-|x|)) : stable for all x
  float t = __expf(-fabsf(x));
  return fmaxf(x, 0.0f) + __logf(1.0f + t);
}

// Streaming pass: per-scale sums of softplus over obj channel (c==4) and
// cls channels (c>=5). acc2[0] += obj_sum, acc2[1] += cls_sum.
__global__ __launch_bounds__(256) void yolo_softplus_sum(
    const float* __restrict__ pred, unsigned n4, float* __restrict__ acc2) {
  unsigned tid = blockIdx.x * blockDim.x + threadIdx.x;
  unsigned stride = gridDim.x * blockDim.x;
  const v4f* __restrict__ p4 = (const v4f*)pred;

  float t_acc = 0.0f;  // softplus sum over channels >= 4
  float o_acc = 0.0f;  // softplus sum over channel == 4
  for (unsigned i = tid; i < n4; i += stride) {
    v4f v = __builtin_nontemporal_load(&p4[i]);  // global_load_b128, th:NT
    unsigned ch = (i * 4u) % 85u;
#pragma unroll
    for (int j = 0; j < 4; ++j) {
      float sp = softplus_f(v[j]);
      unsigned c = ch + (unsigned)j;
      if (c >= 85u) c -= 85u;
      t_acc += (c >= 4u) ? sp : 0.0f;
      o_acc += (c == 4u) ? sp : 0.0f;
    }
  }
  float objA = o_acc;
  float clsA = t_acc - o_acc;

  unsigned lane = threadIdx.x & 31u;
  float partial;
#if __has_builtin(__builtin_amdgcn_wmma_f32_16x16x4_f32)
  // One WMMA folds lane n with lane n+16 for BOTH accumulators and routes
  // obj sums to lanes 0..15 (D row 0) and cls sums to lanes 16..31 (D row 8).
  //   A(16x4): row0 = (1,1,0,0)  -> lane0  a={1,1}, lane16 a={0,0}
  //            row8 = (0,0,1,1)  -> lane8  a={0,0}, lane24 a={1,1}
  //   B(4x16): VGPR0 = objA (K0: lanes0-15, K1: lanes16-31)
  //            VGPR1 = clsA (K2, K3)
  v2f a;
  a.x = (lane == 0u || lane == 24u) ? 1.0f : 0.0f;
  a.y = a.x;
  v2f b;
  b.x = objA;
  b.y = clsA;
  v8f c = {};
  c = __builtin_amdgcn_wmma_f32_16x16x4_f32(false, a, false, b, (short)0, c,
                                            false, false);
  partial = c[0];  // lanes 0-15: obj[n]+obj[n+16]; lanes 16-31: cls[n]+cls[n+16]
#pragma unroll
  for (int off = 8; off > 0; off >>= 1)
    partial += __shfl_xor(partial, off, 32);
#else
  float po = objA, pc = clsA;
#pragma unroll
  for (int off = 16; off > 0; off >>= 1) {
    po += __shfl_xor(po, off, 32);
    pc += __shfl_xor(pc, off, 32);
  }
  partial = (lane < 16u) ? po : pc;
#endif
  // lane 0 now holds the wave's obj sum, lane 16 the wave's cls sum.
  __shared__ float sAcc[2];
  if (threadIdx.x < 2) sAcc[threadIdx.x] = 0.0f;
  __syncthreads();
  if (lane == 0u) atomicAdd(&sAcc[0], partial);
  if (lane == 16u) atomicAdd(&sAcc[1], partial);
  __syncthreads();
  if (threadIdx.x == 0) {
    unsafeAtomicAdd(&acc2[0], sAcc[0]);
    unsafeAtomicAdd(&acc2[1], sAcc[1]);
  }
}

// Correction: subtract logits at target-hit positions (scatter-max semantics:
// each unique (b,gy,gx) / (b,gy,gx,cl) counts exactly once). N<=200 -> O(N^2).
__global__ __launch_bounds__(256) void yolo_correction(
    const float* __restrict__ p0, const float* __restrict__ p1,
    const float* __restrict__ p2, const float* __restrict__ tgt, int nT,
    float* __restrict__ acc) {
  int i = blockIdx.x * blockDim.x + threadIdx.x;
  if (i >= nT) return;
  float cf = tgt[6 * i + 1];
  if (!(cf < (float)YC)) return;  // invalid target: val==0, scatter is no-op
  int b = (int)tgt[6 * i + 0];
  int cl = (int)cf;
  float x = tgt[6 * i + 2];
  float y = tgt[6 * i + 3];

  const float* preds[3] = {p0, p1, p2};
  const int Ws[3] = {80, 40, 20};
#pragma unroll
  for (int s = 0; s < 3; ++s) {
    int W = Ws[s];
    int gx = min((int)floorf(x * (float)W), W - 1);
    int gy = min((int)floorf(y * (float)W), W - 1);
    bool firstObj = true, firstCls = true;
    for (int j = 0; j < i; ++j) {
      float cjf = tgt[6 * j + 1];
      if (!(cjf < (float)YC)) continue;
      int bj = (int)tgt[6 * j + 0];
      int gxj = min((int)floorf(tgt[6 * j + 2] * (float)W), W - 1);
      int gyj = min((int)floorf(tgt[6 * j + 3] * (float)W), W - 1);
      if (bj == b && gxj == gx && gyj == gy) {
        firstObj = false;
        if ((int)cjf == cl) firstCls = false;
      }
    }
    // anchor index is always 0 in the reference scatter
    long long base =
        ((((long long)b * YA + 0) * W + gy) * (long long)W + gx) * YCH;
    if (firstObj) unsafeAtomicAdd(&acc[2 * s + 0], -preds[s][base + 4]);
    if (firstCls) unsafeAtomicAdd(&acc[2 * s + 1], -preds[s][base + 5 + cl]);
  }
}

__global__ void yolo_finalize(const float* __restrict__ acc,
                              float* __restrict__ out) {
  if (threadIdx.x == 0 && blockIdx.x == 0) {
    const float BA = (float)(YB * YA);
    float obj = acc[0] / (BA * 80.0f * 80.0f) + acc[2] / (BA * 40.0f * 40.0f) +
                acc[4] / (BA * 20.0f * 20.0f);
    float cls = acc[1] / (BA * 80.0f * 80.0f * (float)YC) +
                acc[3] / (BA * 40.0f * 40.0f * (float)YC) +
                acc[5] / (BA * 20.0f * 20.0f * (float)YC);
    float total = 1.0f * obj + 0.5f * cls;  // BOX_W * 0 == 0
    out[0] = total;
    out[1] = 0.0f;
    out[2] = obj;
    out[3] = cls;
  }
}

static inline unsigned blocks_for(unsigned n4) {
  unsigned b = (n4 + 2047u) / 2048u;  // ~8 float4 iters per thread
  if (b > 8192u) b = 8192u;
  if (b == 0u) b = 1u;
  return b;
}

extern "C" void kernel_launch(void* const* d_in, const int* in_sizes, int n_in,
                              void* d_out, int out_size, void* d_ws,
                              size_t ws_size, hipStream_t stream) {
  const float* p0 = (const float*)d_in[0];
  const float* p1 = (const float*)d_in[1];
  const float* p2 = (const float*)d_in[2];
  const float* tg = (const float*)d_in[3];
  float* out = (float*)d_out;
  float* acc = (float*)d_ws;  // acc[0..5] = {obj,cls} per scale
  int nT = in_sizes[3] / 6;

  hipMemsetAsync(d_ws, 0, 8 * sizeof(float), stream);

  const float* preds[3] = {p0, p1, p2};
  const unsigned n4s[3] = {(unsigned)in_sizes[0] / 4u,
                           (unsigned)in_sizes[1] / 4u,
                           (unsigned)in_sizes[2] / 4u};
  for (int s = 0; s < 3; ++s) {
    yolo_softplus_sum<<<blocks_for(n4s[s]), 256, 0, stream>>>(
        preds[s], n4s[s], acc + 2 * s);
  }
  yolo_correction<<<(nT + 255) / 256, 256, 0, stream>>>(p0, p1, p2, tg, nT,
                                                        acc);
  yolo_finalize<<<1, 32, 0, stream>>>(acc, out);
}